// AttentionLayer_51178830299318
// MI455X (gfx1250) — compile-verified
//
#include <hip/hip_runtime.h>
#include <math.h>

#define BB 16
#define SS 8192
#define DD 512
#define HH 8
#define HD 64
#define NS 3
#define NSPLIT 8

typedef __attribute__((ext_vector_type(2))) float v2f;
typedef __attribute__((ext_vector_type(8))) float v8f;

__device__ __forceinline__ v8f wmma4(v2f a, v2f b, v8f c) {
  // V_WMMA_F32_16X16X4_F32 : D = A(16x4 f32) x B(4x16 f32) + C(16x16 f32)
  return __builtin_amdgcn_wmma_f32_16x16x4_f32(false, a, false, b, (short)0, c,
                                               false, false);
}

// Non-temporal f32/f32x2 loads: encoder stream (268 MB) must not thrash the
// 192 MB L2 that holds the reused scores/weights working set.
__device__ __forceinline__ v2f nt_load2(const float* p) {
  return __builtin_nontemporal_load((const v2f*)p);
}
__device__ __forceinline__ float nt_load1(const float* p) {
  return __builtin_nontemporal_load(p);
}

// ---------------- workspace layout (floats) ----------------
#define OFF_Q    0                                   // B*D            = 8192
#define OFF_QKP  (OFF_Q + BB*DD)                     // NS*B*8*D       = 196608
#define OFF_QKB  (OFF_QKP + NS*BB*HH*DD)             // NS*B*H         = 384
#define OFF_SC   (OFF_QKB + NS*BB*HH)                // B*H*(S+S/2+S/4)= 1835008
#define SC_TOT   (BB*HH*(SS + SS/2 + SS/4))
#define OFF_CTX  (OFF_SC + SC_TOT)                   // NS*B*H*D       = 196608
#define OFF_COMB (OFF_CTX + NS*BB*HH*DD)             // B*D            = 8192

// ---------------- 1) q = dh @ Wq + bq ----------------
__global__ void k_qproj(const float* __restrict__ dh, const float* __restrict__ Wq,
                        const float* __restrict__ bq, float* __restrict__ q) {
  __shared__ float s[DD];
  int b = blockIdx.x, d = threadIdx.x;
  s[d] = dh[b * DD + d];
  __syncthreads();
  float acc = bq[d];
#pragma unroll 4
  for (int c = 0; c < DD; ++c) acc = fmaf(s[c], Wq[(size_t)c * DD + d], acc);
  q[b * DD + d] = acc;
}

// ---------------- 2) fold q into Wk: qkp[i][b][h][c], qkb[i][b][h] ----------------
__global__ void k_qk(const float* __restrict__ q, const float* __restrict__ Wk,
                     const float* __restrict__ bk, float* __restrict__ qkp,
                     float* __restrict__ qkb) {
  int i = blockIdx.x / BB, b = blockIdx.x % BB;
  int c = threadIdx.x;
  __shared__ float sq[DD];
  sq[c] = q[b * DD + c];
  __syncthreads();
  const float* wrow = Wk + ((size_t)i * DD + c) * DD;   // row c of Wk[i]
  float* outb = qkp + (size_t)(i * BB + b) * HH * DD;
#pragma unroll
  for (int h = 0; h < HH; ++h) {
    float acc = 0.f;
#pragma unroll 8
    for (int d = 0; d < HD; ++d) acc = fmaf(sq[h * HD + d], wrow[h * HD + d], acc);
    outb[h * DD + c] = acc;
  }
  if (c < HH) {
    float acc = 0.f;
    const float* bkr = bk + i * DD + c * HD;
    for (int d = 0; d < HD; ++d) acc = fmaf(sq[c * HD + d], bkr[d], acc);
    qkb[(i * BB + b) * HH + c] = acc;
  }
}

// ---------------- 3) scores[i][b][h][s] = qk . kv  (WMMA f32, K=512) ----------------
__global__ void k_scores(const float* __restrict__ enc, const float* __restrict__ qkp,
                         const float* __restrict__ qkb, float* __restrict__ scores) {
  constexpr int DIL[NS] = {1, 2, 4};
  constexpr int SI[NS] = {SS, SS / 2, SS / 4};
  constexpr size_t SCB[NS] = {0, (size_t)BB * HH * SS,
                              (size_t)BB * HH * SS + (size_t)BB * HH * (SS / 2)};
  int x = blockIdx.x;  // 0..111 s-chunks across scales (128 s per block)
  int b = blockIdx.y;
  int i, chunk;
  if (x < 64)      { i = 0; chunk = x; }
  else if (x < 96) { i = 1; chunk = x - 64; }
  else             { i = 2; chunk = x - 96; }

  __shared__ float sA[HH * DD];  // 16 KB folded-query tile (8 heads x 512)
  const float* abase = qkp + (size_t)(i * BB + b) * HH * DD;
  for (int idx = threadIdx.x; idx < HH * DD; idx += 256) sA[idx] = abase[idx];
  __syncthreads();

  int wave = threadIdx.x >> 5;
  int lane = threadIdx.x & 31;
  int half = lane >> 4;
  int l16 = lane & 15;
  int d0 = half * 2;

  int s = chunk * 128 + wave * 16 + l16;  // logical dilated index (column n)
  const float* brow = enc + ((size_t)b * SS + (size_t)s * DIL[i]) * DD;
  const float* arow = sA + (size_t)(l16 & 7) * DD;  // rows 8..15 alias (discarded)

  v8f acc0 = {}, acc1 = {};
#pragma unroll 4
  for (int kk = 0; kk < DD / 4; kk += 2) {
    v2f a0 = *(const v2f*)(arow + 4 * kk + d0);
    v2f b0 = nt_load2(brow + 4 * kk + d0);
    acc0 = wmma4(a0, b0, acc0);
    v2f a1 = *(const v2f*)(arow + 4 * kk + 4 + d0);
    v2f b1 = nt_load2(brow + 4 * kk + 4 + d0);
    acc1 = wmma4(a1, b1, acc1);
  }
  v8f acc = acc0 + acc1;

  if (half == 0) {  // lanes 0..15 hold heads 0..7, column s
    float* orow = scores + SCB[i] + (size_t)b * HH * SI[i];
    const float* qb = qkb + (i * BB + b) * HH;
#pragma unroll
    for (int r = 0; r < HH; ++r)
      orow[(size_t)r * SI[i] + s] = acc[r] * 0.125f + qb[r];
  }
}

// ---------------- 4) row softmax over s ----------------
__global__ void k_softmax(float* __restrict__ scores) {
  constexpr int SI[NS] = {SS, SS / 2, SS / 4};
  constexpr size_t SCB[NS] = {0, (size_t)BB * HH * SS,
                              (size_t)BB * HH * SS + (size_t)BB * HH * (SS / 2)};
  int r = blockIdx.x;      // NS*B*H = 384 rows
  int i = r >> 7;          // /128
  int rem = r & 127;       // b*H + h
  int n = SI[i];
  float* row = scores + SCB[i] + (size_t)rem * n;
  __shared__ float red[256];
  int t = threadIdx.x;
  float m = -3.402823466e38f;
  for (int s = t; s < n; s += 256) m = fmaxf(m, row[s]);
  red[t] = m;
  __syncthreads();
  for (int o = 128; o > 0; o >>= 1) {
    if (t < o) red[t] = fmaxf(red[t], red[t + o]);
    __syncthreads();
  }
  m = red[0];
  __syncthreads();
  float sum = 0.f;
  for (int s = t; s < n; s += 256) {
    float e = expf(row[s] - m);
    row[s] = e;
    sum += e;
  }
  red[t] = sum;
  __syncthreads();
  for (int o = 128; o > 0; o >>= 1) {
    if (t < o) red[t] += red[t + o];
    __syncthreads();
  }
  float inv = 1.f / red[0];
  for (int s = t; s < n; s += 256) row[s] *= inv;
}

// ---------------- 5) ctx[i][b][h][c] = sum_s w * kv  (WMMA f32, K=S, split 8x) ----------------
__global__ void k_ctx(const float* __restrict__ enc, const float* __restrict__ scores,
                      float* __restrict__ ctx) {
  constexpr int DIL[NS] = {1, 2, 4};
  constexpr int SI[NS] = {SS, SS / 2, SS / 4};
  constexpr size_t SCB[NS] = {0, (size_t)BB * HH * SS,
                              (size_t)BB * HH * SS + (size_t)BB * HH * (SS / 2)};
  int x = blockIdx.x;  // NS*B*NSPLIT = 384
  int i = x / (BB * NSPLIT);
  int b = (x / NSPLIT) % BB;
  int sp = x % NSPLIT;
  int n = SI[i];
  int ch = n / NSPLIT;                       // 1024 / 512 / 256
  int lg = (i == 0) ? 10 : ((i == 1) ? 9 : 8);
  int s0 = sp * ch;

  __shared__ float sW[HH * (SS / NSPLIT)];   // 32 KB softmax-weight chunk
  const float* wbase = scores + SCB[i] + (size_t)b * HH * n;
  for (int idx = threadIdx.x; idx < HH * ch; idx += 256) {
    int h = idx >> lg;
    int s = idx & (ch - 1);
    sW[(size_t)h * ch + s] = wbase[(size_t)h * n + s0 + s];
  }
  __syncthreads();

  int wave = threadIdx.x >> 5;
  int lane = threadIdx.x & 31;
  int half = lane >> 4;
  int l16 = lane & 15;
  int d0 = half * 2;
  const float* wrow = sW + (size_t)(l16 & 7) * ch;  // head rows (>=8 alias, discarded)
  const float* eb = enc + (size_t)b * SS * DD;

  for (int ct = wave; ct < 32; ct += 8) {    // 32 column tiles of 16 channels
    int c = ct * 16 + l16;                   // column n
    v8f acc0 = {}, acc1 = {};
#pragma unroll 4
    for (int kk = 0; kk < ch / 4; kk += 2) {
      v2f a0 = *(const v2f*)(wrow + 4 * kk + d0);
      int sl0 = s0 + 4 * kk + d0;
      v2f b0 = {nt_load1(&eb[(size_t)sl0 * DIL[i] * DD + c]),
                nt_load1(&eb[(size_t)(sl0 + 1) * DIL[i] * DD + c])};
      acc0 = wmma4(a0, b0, acc0);
      v2f a1 = *(const v2f*)(wrow + 4 * kk + 4 + d0);
      int sl1 = sl0 + 4;
      v2f b1 = {nt_load1(&eb[(size_t)sl1 * DIL[i] * DD + c]),
                nt_load1(&eb[(size_t)(sl1 + 1) * DIL[i] * DD + c])};
      acc1 = wmma4(a1, b1, acc1);
    }
    v8f acc = acc0 + acc1;
    if (half == 0) {
#pragma unroll
      for (int r = 0; r < HH; ++r)
        atomicAdd(&ctx[((size_t)(i * BB + b) * HH + r) * DD + c], acc[r]);
    }
  }
}

// ---------------- 6) combined = sum_i sw_i * (ctx_i @ Wv_i + bv_i) ----------------
__global__ void k_av(const float* __restrict__ ctx, const float* __restrict__ Wv,
                     const float* __restrict__ bv, const float* __restrict__ swin,
                     float* __restrict__ comb) {
  int b = blockIdx.y;
  int d = blockIdx.x * 256 + threadIdx.x;
  int h = d >> 6;
  float w0 = swin[0], w1 = swin[1], w2 = swin[2];
  float mx = fmaxf(w0, fmaxf(w1, w2));
  float e0 = expf(w0 - mx), e1 = expf(w1 - mx), e2 = expf(w2 - mx);
  float invs = 1.f / (e0 + e1 + e2);
  float sw[NS] = {e0 * invs, e1 * invs, e2 * invs};
  float acc = 0.f;
  for (int i = 0; i < NS; ++i) {
    const float* crow = ctx + ((size_t)(i * BB + b) * HH + h) * DD;
    const float* wcol = Wv + (size_t)i * DD * DD + d;
    float a = bv[i * DD + d];
#pragma unroll 4
    for (int c = 0; c < DD; ++c) a = fmaf(crow[c], wcol[(size_t)c * DD], a);
    acc = fmaf(sw[i], a, acc);
  }
  comb[(size_t)b * DD + d] = acc;
}

// ---------------- 7) out = combined @ Wo + bo ----------------
__global__ void k_out(const float* __restrict__ comb, const float* __restrict__ Wo,
                      const float* __restrict__ bo, float* __restrict__ out) {
  __shared__ float s[DD];
  int b = blockIdx.x, d = threadIdx.x;
  s[d] = comb[b * DD + d];
  __syncthreads();
  float acc = bo[d];
#pragma unroll 4
  for (int c = 0; c < DD; ++c) acc = fmaf(s[c], Wo[(size_t)c * DD + d], acc);
  out[(size_t)b * DD + d] = acc;
}

extern "C" void kernel_launch(void* const* d_in, const int* in_sizes, int n_in,
                              void* d_out, int out_size, void* d_ws, size_t ws_size,
                              hipStream_t stream) {
  const float* dh  = (const float*)d_in[0];
  const float* enc = (const float*)d_in[1];
  const float* Wq  = (const float*)d_in[2];
  const float* bq  = (const float*)d_in[3];
  const float* Wk  = (const float*)d_in[4];
  const float* bk  = (const float*)d_in[5];
  const float* Wv  = (const float*)d_in[6];
  const float* bv  = (const float*)d_in[7];
  const float* sws = (const float*)d_in[8];
  const float* Wo  = (const float*)d_in[9];
  const float* bo  = (const float*)d_in[10];
  float* out = (float*)d_out;
  float* ws = (float*)d_ws;

  float* q    = ws + OFF_Q;
  float* qkp  = ws + OFF_QKP;
  float* qkb  = ws + OFF_QKB;
  float* sc   = ws + OFF_SC;
  float* ctx  = ws + OFF_CTX;
  float* comb = ws + OFF_COMB;

  hipMemsetAsync(ctx, 0, (size_t)NS * BB * HH * DD * sizeof(float), stream);

  k_qproj<<<BB, DD, 0, stream>>>(dh, Wq, bq, q);
  k_qk<<<NS * BB, DD, 0, stream>>>(q, Wk, bk, qkp, qkb);
  dim3 g3(112, BB);
  k_scores<<<g3, 256, 0, stream>>>(enc, qkp, qkb, sc);
  k_softmax<<<NS * BB * HH, 256, 0, stream>>>(sc);
  k_ctx<<<NS * BB * NSPLIT, 256, 0, stream>>>(enc, sc, ctx);
  dim3 g6(2, BB);
  k_av<<<g6, 256, 0, stream>>>(ctx, Wv, bv, sws, comb);
  k_out<<<BB, DD, 0, stream>>>(comb, Wo, bo, out);
}